// NestedNerTagger_7619271983324
// MI455X (gfx1250) — compile-verified
//
#include <hip/hip_runtime.h>
#include <hip/hip_bf16.h>
#include <math.h>

// Problem constants (match reference)
#define BB        64
#define SS        512
#define HH        768
#define NUM_BD    5
#define N_CLASSES 9
#define N_REGIONS 4096
#define GAMMA_F   0.3f

typedef __attribute__((ext_vector_type(2))) float v2f;
typedef __attribute__((ext_vector_type(8))) float v8f;

// Padded-B LDS layout: entry for (k, n) lives at (k>>1)*32 + 2*n + (k&1),
// so the per-lane WMMA B fragment {W[k][n], W[k+1][n]} (k even) is one
// 8B-aligned contiguous pair -> single conflict-free ds_load_b64.
#define PADB_DWORDS ((HH / 2) * 32)   // 12288 floats = 48 KB

// ---------------------------------------------------------------------------
// Init: zero the three accumulators in workspace:
//   acc[0] = sum(seg * nll_bd), acc[1] = sum(seg), acc[2] = sum(nll_region)
// ---------------------------------------------------------------------------
__global__ void nnt_init_kernel(float* acc) {
    if (threadIdx.x < 3) acc[threadIdx.x] = 0.0f;
}

// ---------------------------------------------------------------------------
// Kernel 1: sentence-boundary logits via V_WMMA_F32_16X16X4_F32 + masked NLL.
// One wave handles a 16-token x 16-class tile (classes 5..15 zero-padded in
// LDS, so the K-loop is branch-free). Grid: 512 x 128 -> 2048 tiles.
// ---------------------------------------------------------------------------
__global__ void __launch_bounds__(128)
nnt_bd_kernel(const float* __restrict__ word,   // [B*S, H]
              const float* __restrict__ Ws,     // [H, 5]
              const float* __restrict__ bs,     // [5]
              const int*   __restrict__ bd_tgt, // [B*S]
              const int*   __restrict__ seg,    // [B*S]
              float* __restrict__ acc) {
    __shared__ float padB[PADB_DWORDS];
    __shared__ float tile[4][16 * 16];

    // ---- Stage zero-padded W_s into LDS (pair-contiguous layout). ----
    for (int idx = threadIdx.x; idx < HH * 16; idx += 128) {
        const int k = idx >> 4;
        const int n = idx & 15;
        padB[(k >> 1) * 32 + (n << 1) + (k & 1)] =
            (n < NUM_BD) ? Ws[k * NUM_BD + n] : 0.0f;
    }
    __syncthreads();

    const int wave = threadIdx.x >> 5;
    const int lane = threadIdx.x & 31;
    const int n    = lane & 15;        // column (class) for B/C/D fragments
    const int hi   = lane >> 4;        // half-wave: K offset +2, M offset +8
    const int rowBase = (blockIdx.x * 4 + wave) * 16;

    const float* arow = word + (size_t)(rowBase + n) * HH;  // A row M = lane&15
    const float* bbase = &padB[(hi << 5) + (n << 1)];       // + (k0>>1)*32

    v8f c = {};
    for (int k0 = 0; k0 < HH; k0 += 4) {
        const int kk = k0 + 2 * hi;
        v2f a = *(const v2f*)(arow + kk);                 // global_load_b64
        v2f b = *(const v2f*)(bbase + (k0 >> 1) * 32);    // ds_load_b64
        c = __builtin_amdgcn_wmma_f32_16x16x4_f32(false, a, false, b,
                                                  (short)0, c, false, false);
    }

    // Spill D tile to LDS: lane holds column n, rows j + 8*hi
    float* t = tile[wave];
#pragma unroll
    for (int j = 0; j < 8; ++j) t[(j + 8 * hi) * 16 + n] = c[j];
    __syncthreads();

    // Per-token 5-way log-softmax NLL, masked by seg
    float part_num = 0.0f, part_den = 0.0f;
    if (lane < 16) {
        const int tok = rowBase + lane;
        float x[NUM_BD];
        float mx = -3.4e38f;
#pragma unroll
        for (int k = 0; k < NUM_BD; ++k) {
            x[k] = t[lane * 16 + k] + bs[k];
            mx = fmaxf(mx, x[k]);
        }
        float se = 0.0f;
#pragma unroll
        for (int k = 0; k < NUM_BD; ++k) se += __expf(x[k] - mx);
        const float lse = mx + __logf(se);
        const int   tg  = bd_tgt[tok];
        const float m   = (float)seg[tok];
        part_num = m * (lse - x[tg]);
        part_den = m;
    }
#pragma unroll
    for (int off = 16; off >= 1; off >>= 1) {
        part_num += __shfl_down(part_num, off, 32);
        part_den += __shfl_down(part_den, off, 32);
    }
    if (lane == 0) {
        atomicAdd(&acc[0], part_num);
        atomicAdd(&acc[1], part_den);
    }
}

// ---------------------------------------------------------------------------
// Kernel 2: region mean-pool into LDS, then the same WMMA f32 16x16x4 K-loop
// for the [16 x 9] region-classification tile, then 9-way NLL.
// Grid: 256 blocks x 256 threads; each block owns 16 regions.
// ---------------------------------------------------------------------------
#define SM_STRIDE 772  // 768 + 4 dwords: stride % 64 == 4 -> conflict-free b64

__global__ void __launch_bounds__(256)
nnt_region_kernel(const float* __restrict__ word,   // [B*S, H]
                  const float* __restrict__ Wr,     // [H, 9]
                  const float* __restrict__ br,     // [9]
                  const int*   __restrict__ rstart, // [R]
                  const int*   __restrict__ rlen,   // [R]
                  const int*   __restrict__ rbat,   // [R]
                  const int*   __restrict__ rlab,   // [R]
                  float* __restrict__ acc) {
    __shared__ float smean[16 * SM_STRIDE];  // 16 region mean vectors
    __shared__ float padB[PADB_DWORDS];      // zero-padded W_r
    __shared__ float tile[16 * 16];          // logits tile

    const int tid = threadIdx.x;

    // ---- Stage zero-padded W_r into LDS (pair-contiguous layout). ----
    for (int idx = tid; idx < HH * 16; idx += 256) {
        const int k = idx >> 4;
        const int n = idx & 15;
        padB[(k >> 1) * 32 + (n << 1) + (k & 1)] =
            (n < N_CLASSES) ? Wr[k * N_CLASSES + n] : 0.0f;
    }

    // ---- Phase 1: pool means. 16 threads per region, coalesced over H. ----
    {
        const int r    = tid >> 4;           // local region 0..15
        const int lsub = tid & 15;
        const int reg  = blockIdx.x * 16 + r;
        const int len  = rlen[reg] + 1;      // reference divides by length+1
        const float inv = 1.0f / (float)len;
        const float* base = word + ((size_t)rbat[reg] * SS + rstart[reg]) * HH;
        for (int h = lsub; h < HH; h += 16) {
            float s = 0.0f;
            for (int row = 0; row < len; ++row) s += base[(size_t)row * HH + h];
            smean[r * SM_STRIDE + h] = s * inv;
        }
    }
    __syncthreads();

    // ---- Phase 2: wave 0 runs the WMMA K-loop entirely out of LDS. ----
    if (tid < 32) {
        const int lane = tid;
        const int n    = lane & 15;
        const int hi   = lane >> 4;
        const float* amean = &smean[(lane & 15) * SM_STRIDE + 2 * hi];
        const float* bbase = &padB[(hi << 5) + (n << 1)];

        v8f c = {};
        for (int k0 = 0; k0 < HH; k0 += 4) {
            v2f a = *(const v2f*)(amean + k0);             // ds_load_b64
            v2f b = *(const v2f*)(bbase + (k0 >> 1) * 32); // ds_load_b64
            c = __builtin_amdgcn_wmma_f32_16x16x4_f32(false, a, false, b,
                                                      (short)0, c, false, false);
        }
#pragma unroll
        for (int j = 0; j < 8; ++j) tile[(j + 8 * hi) * 16 + n] = c[j];
    }
    __syncthreads();

    // ---- Phase 3: per-region 9-way log-softmax NLL. ----
    if (tid < 16) {
        const int reg = blockIdx.x * 16 + tid;
        float x[N_CLASSES];
        float mx = -3.4e38f;
#pragma unroll
        for (int k = 0; k < N_CLASSES; ++k) {
            x[k] = tile[tid * 16 + k] + br[k];
            mx = fmaxf(mx, x[k]);
        }
        float se = 0.0f;
#pragma unroll
        for (int k = 0; k < N_CLASSES; ++k) se += __expf(x[k] - mx);
        const float lse = mx + __logf(se);
        atomicAdd(&acc[2], lse - x[rlab[reg]]);
    }
}

// ---------------------------------------------------------------------------
// Finalize: blend the two mean losses into the scalar output.
// ---------------------------------------------------------------------------
__global__ void nnt_finalize_kernel(const float* __restrict__ acc,
                                    float* __restrict__ out) {
    if (threadIdx.x == 0) {
        const float ent = acc[2] / (float)N_REGIONS;
        const float bd  = acc[0] / (acc[1] + 1e-6f);
        out[0] = GAMMA_F * ent + (1.0f - GAMMA_F) * bd;
    }
}

// ---------------------------------------------------------------------------
extern "C" void kernel_launch(void* const* d_in, const int* in_sizes, int n_in,
                              void* d_out, int out_size, void* d_ws, size_t ws_size,
                              hipStream_t stream) {
    const float* word   = (const float*)d_in[0];
    const float* Ws     = (const float*)d_in[1];
    const float* bs     = (const float*)d_in[2];
    const float* Wr     = (const float*)d_in[3];
    const float* br     = (const float*)d_in[4];
    const int*   bd_tgt = (const int*)d_in[5];
    const int*   seg    = (const int*)d_in[6];
    const int*   rstart = (const int*)d_in[7];
    const int*   rlen   = (const int*)d_in[8];
    const int*   rbat   = (const int*)d_in[9];
    const int*   rlab   = (const int*)d_in[10];

    float* acc = (float*)d_ws;
    float* out = (float*)d_out;

    nnt_init_kernel<<<1, 32, 0, stream>>>(acc);
    // 32768 tokens / 16 rows per wave-tile = 2048 tiles; 4 waves per block
    nnt_bd_kernel<<<512, 128, 0, stream>>>(word, Ws, bs, bd_tgt, seg, acc);
    // 4096 regions / 16 per block
    nnt_region_kernel<<<256, 256, 0, stream>>>(word, Wr, br, rstart, rlen,
                                               rbat, rlab, acc);
    nnt_finalize_kernel<<<1, 32, 0, stream>>>(acc, out);
}